// Use_Sensity_40862318854166
// MI455X (gfx1250) — compile-verified
//
#include <hip/hip_runtime.h>
#include <math.h>

// Problem constants from the reference
#define BB 64
#define CC 512
#define HH 56
#define WW 56
#define K9 9
#define KLEN (CC * K9)                    // 4608 = GEMV reduction length
#define PLANE (HH * WW)                   // 3136 floats per (b,c) plane
#define PLANE4 (PLANE / 4)                // 784 float4 per plane
#define OUT_ELEMS ((size_t)BB * CC * HH * WW)  // 102,760,448

typedef __attribute__((ext_vector_type(2))) float v2f;
typedef __attribute__((ext_vector_type(4))) float v4f;
typedef __attribute__((ext_vector_type(8))) float v8f;

// ---------------------------------------------------------------------------
// Kernel 1: s[i] = sum_b ssty[b*4608 + i], i in [0,4608)
// ---------------------------------------------------------------------------
__global__ void sum_ssty_kernel(const float* __restrict__ ssty,
                                float* __restrict__ s) {
    int i = blockIdx.x * blockDim.x + threadIdx.x;
    if (i >= KLEN) return;
    float acc = 0.0f;
    #pragma unroll 8
    for (int b = 0; b < BB; ++b) {
        acc += ssty[(size_t)b * KLEN + i];   // coalesced across threads
    }
    s[i] = acc;
}

// ---------------------------------------------------------------------------
// Kernel 2: GEMV via V_WMMA_F32_16X16X4_F32 + bias + sigmoid(beta * m)
// One wave (32 threads) per tile of 16 output channels; 32 blocks total.
//
// A (16x4 f32, per ISA layout): lanes 0-15 hold row M=lane, VGPR0/1 = K=k,k+1;
//                               lanes 16-31 hold row M=lane-16, K=k+2,k+3.
// B (4x16 f32): VGPR v, lanes 0-15 -> K=v (N=lane), lanes 16-31 -> K=v+2.
//               We broadcast s[k..k+3] across all 16 columns, so every
//               column of D holds the same dot product; read column N=0.
// D (16x16 f32): VGPR v: lanes 0-15 = M=v, lanes 16-31 = M=v+8.
// ---------------------------------------------------------------------------
__global__ __launch_bounds__(32)
void mask_wmma_kernel(const float* __restrict__ s,
                      const float* __restrict__ w,       // [512, 4608]
                      const float* __restrict__ bias,    // [512]
                      const float* __restrict__ beta,    // [1]
                      float* __restrict__ mask_out) {    // [512]
    const int lane  = threadIdx.x;          // 0..31
    const int tile  = blockIdx.x;           // 0..31
    const int m     = tile * 16 + (lane & 15);
    const int khalf = (lane >> 4) * 2;      // 0 for lanes 0-15, 2 for 16-31

    const float* wrow = w + (size_t)m * KLEN + khalf;
    const float* svec = s + khalf;

    v8f acc = {};
    for (int k = 0; k < KLEN; k += 4) {
        v2f a = *(const v2f*)(wrow + k);    // W[m][k+khalf], W[m][k+khalf+1]
        v2f b = *(const v2f*)(svec + k);    // s[k+khalf],   s[k+khalf+1]
        // 8 args: (neg_a, A, neg_b, B, c_mod, C, reuse_a, reuse_b)
        acc = __builtin_amdgcn_wmma_f32_16x16x4_f32(
            false, a, false, b, (short)0, acc, false, false);
    }

    // Column N=0 lives in lane 0 (M=v) and lane 16 (M=v+8)
    if ((lane & 15) == 0) {
        const int mbase = tile * 16 + (lane >> 4) * 8;
        const float bt = beta[0];
        #pragma unroll
        for (int v = 0; v < 8; ++v) {
            float z = (acc[v] + bias[mbase + v]) * bt;
            mask_out[mbase + v] = 1.0f / (1.0f + __expf(-z));
        }
    }
}

// ---------------------------------------------------------------------------
// Kernel 3: out = x * mask[c], one block per (b,c) plane.
// mask[c] is block-uniform -> scalar load; float4 (B128) non-temporal
// streaming (411 MB in + 411 MB out, touched once; bigger than 192 MB L2).
// ---------------------------------------------------------------------------
__global__ __launch_bounds__(256)
void scale_kernel(const v4f* __restrict__ x,
                  const float* __restrict__ mask,
                  v4f* __restrict__ out) {
    const int plane = blockIdx.x;           // b*C + c
    const int c     = plane & (CC - 1);     // C = 512, power of two
    const float mval = mask[c];
    const size_t base = (size_t)plane * PLANE4;

    for (int j = threadIdx.x; j < PLANE4; j += 256) {
        v4f v = __builtin_nontemporal_load(&x[base + j]);
        v *= mval;
        __builtin_nontemporal_store(v, &out[base + j]);
    }
}

// ---------------------------------------------------------------------------
extern "C" void kernel_launch(void* const* d_in, const int* in_sizes, int n_in,
                              void* d_out, int out_size, void* d_ws, size_t ws_size,
                              hipStream_t stream) {
    const float* x      = (const float*)d_in[0];   // [64,512,56,56]
    const float* ssty   = (const float*)d_in[1];   // [64,512,3,3]
    const float* beta   = (const float*)d_in[2];   // [1]
    const float* conv_w = (const float*)d_in[3];   // [512,512,3,3]
    const float* conv_b = (const float*)d_in[4];   // [512]

    float* out_main = (float*)d_out;               // [64*512*56*56]
    float* out_mask = (float*)d_out + OUT_ELEMS;   // [512]
    float* s        = (float*)d_ws;                // 4608 floats scratch

    // 1) batch-reduce ssty -> s[4608]
    sum_ssty_kernel<<<(KLEN + 255) / 256, 256, 0, stream>>>(ssty, s);

    // 2) f32 WMMA GEMV + bias + sigmoid -> mask[512]
    mask_wmma_kernel<<<CC / 16, 32, 0, stream>>>(s, conv_w, conv_b, beta, out_mask);

    // 3) broadcast scale (bandwidth-bound streaming)
    scale_kernel<<<BB * CC, 256, 0, stream>>>((const v4f*)x, out_mask,
                                              (v4f*)out_main);
}